// RGCN_53833120088188
// MI455X (gfx1250) — compile-verified
//
#include <hip/hip_runtime.h>

typedef float v2f __attribute__((ext_vector_type(2)));
typedef float v4f __attribute__((ext_vector_type(4)));
typedef float v8f __attribute__((ext_vector_type(8)));

#define RGCN_D 128
#define RGCN_H 128
#define RGCN_R 8
#define RGCN_C 16
#define XT_STRIDE ((RGCN_R + 1) * RGCN_H) /* 1152 floats per node in xt scratch */

// LDS B-tile geometry: per K-chunk (4 K-rows) we store two "halves"
// (K+0/K+1 pairs for lanes 0-15, K+2/K+3 pairs for lanes 16-31).
// Each half holds 128 cols x 2 floats = 256 floats, padded to 288 so the
// two lane-halves hit disjoint LDS bank ranges (288 mod 64 == 32).
#define BH_STRIDE 288                 /* floats per half (256 + 32 pad) */
#define BK_STRIDE (2 * BH_STRIDE)     /* floats per K-chunk = 576 */
#define SW_FLOATS (16 * BK_STRIDE)    /* 16 K-chunks per 64-row stage = 9216 (36 KB) */

// ---------------------------------------------------------------------------
// Degree counting / normalization (structure-only, recomputed each call)
// ---------------------------------------------------------------------------
__global__ __launch_bounds__(256) void k_zero_cnt(int* __restrict__ cnt, int n) {
  int i = blockIdx.x * 256 + threadIdx.x;
  if (i < n) cnt[i] = 0;
}

__global__ __launch_bounds__(256) void k_count(const int* __restrict__ dst,
                                               const int* __restrict__ et,
                                               int* __restrict__ cnt, int E) {
  int e = blockIdx.x * 256 + threadIdx.x;
  if (e < E) atomicAdd(&cnt[dst[e] * RGCN_R + et[e]], 1);
}

__global__ __launch_bounds__(256) void k_norm(const int* __restrict__ cnt,
                                              float* __restrict__ norm, int n) {
  int i = blockIdx.x * 256 + threadIdx.x;
  if (i < n) {
    float c = (float)cnt[i];
    norm[i] = 1.0f / fmaxf(c, 1.0f);
  }
}

// ---------------------------------------------------------------------------
// Batched per-relation GEMM on WMMA f32 16x16x4:
//   xt[n, r, :] = X[n, :] @ W_r   (r < R)      xt[n, R, :] = X[n, :] @ root
// grid = (ceil(N/128), R+1); block = 256 (8 waves) covering 128 rows x 128 cols.
// Each wave owns a 32x64 tile (2 row-frags x 4 col-tiles, 8 accumulators):
// halves B-side LDS traffic vs a 16x128 tile at equal WMMA count.
// Weights are LDS-staged pre-swizzled into WMMA B-fragment layout (one aligned
// ds_load_b64 per fragment); B fragments are double-buffered in registers and
// A fragments prefetched in 16-load clauses.
// ---------------------------------------------------------------------------
__global__ __launch_bounds__(256) void k_gemm_rel(const float* __restrict__ X,
                                                  const float* __restrict__ W,
                                                  const float* __restrict__ root,
                                                  float* __restrict__ xt, int N) {
  __shared__ float sW[SW_FLOATS];  // 36 KB: one 64-row K-half, B-frag swizzled

  const int r = blockIdx.y;
  const float* Wr = (r < RGCN_R) ? (W + (size_t)r * RGCN_D * RGCN_H) : root;

  const int tid = threadIdx.x;
  const int lane = tid & 31;
  const int wave = tid >> 5;
  const int halfLane = lane & 15;
  const int kSel = (lane >> 4) << 1;  // lanes 0-15 -> K+0/K+1, lanes 16-31 -> K+2/K+3
  const int rg = wave >> 1;           // row group 0..3 (32 rows each)
  const int cg = wave & 1;            // col group 0..1 (64 cols each)
  const int m0 = blockIdx.x * 128 + rg * 32;
  const int c0 = cg * 64;

  int aRow0 = m0 + halfLane;
  int aRow1 = m0 + 16 + halfLane;
  if (aRow0 >= N) aRow0 = N - 1;  // clamp loads; stores are guarded
  if (aRow1 >= N) aRow1 = N - 1;
  const float* aPtr0 = X + (size_t)aRow0 * RGCN_D;
  const float* aPtr1 = X + (size_t)aRow1 * RGCN_D;

  v8f acc[8];  // [a(2)][ct(4)] -> acc[a*4+ct]
#pragma unroll
  for (int t = 0; t < 8; ++t) acc[t] = v8f{};

  // Lane-invariant base into the swizzled B tile for this lane / col group.
  const float* bBase = &sW[(kSel >> 1) * BH_STRIDE + (c0 + halfLane) * 2];

  for (int kh = 0; kh < 2; ++kh) {
    __syncthreads();  // protect sW from previous half's readers
    {
      // Stage 64 K-rows swizzled: 1024 units of (2 rows x 4 cols); 4 per thread.
      const float* wsrc = Wr + kh * 64 * RGCN_H;
#pragma unroll
      for (int j = 0; j < 4; ++j) {
        const int u = tid + 256 * j;     // 0..1023
        const int k2 = u >> 5;           // pair-of-rows index 0..31
        const int hb = (u & 31) << 2;    // col base 0..124 step 4
        const int k = k2 << 1;
        const v4f w0 = *(const v4f*)(wsrc + k * RGCN_H + hb);
        const v4f w1 = *(const v4f*)(wsrc + (k + 1) * RGCN_H + hb);
        const int kc = k2 >> 1;          // K-chunk
        const int half = k2 & 1;         // which lane-half pair
        float* dstp = &sW[kc * BK_STRIDE + half * BH_STRIDE + hb * 2];
        v4f o0 = {w0.x, w1.x, w0.y, w1.y};
        v4f o1 = {w0.z, w1.z, w0.w, w1.w};
        *(v4f*)(dstp) = o0;
        *(v4f*)(dstp + 4) = o1;
      }
    }
    __syncthreads();

    // Two sub-blocks of 8 K-chunks each (keeps A-prefetch at 32 VGPRs).
#pragma unroll
    for (int sub = 0; sub < 2; ++sub) {
      const int kb0 = kh * 64 + sub * 32 + kSel;  // global K base for this lane
      // Prefetch 8 A fragments per row group (one 16-load clause).
      v2f a0[8], a1[8];
#pragma unroll
      for (int kc = 0; kc < 8; ++kc) {
        a0[kc] = *(const v2f*)(aPtr0 + kb0 + kc * 4);
        a1[kc] = *(const v2f*)(aPtr1 + kb0 + kc * 4);
      }

      const float* bSub = bBase + (sub * 8) * BK_STRIDE;
      // Double-buffered B fragments (4 per K-chunk).
      v2f bf[4];
#pragma unroll
      for (int ct = 0; ct < 4; ++ct)
        bf[ct] = *(const v2f*)(bSub + ct * 32);

#pragma unroll
      for (int kc = 0; kc < 8; ++kc) {
        v2f bn[4];
        if (kc < 7) {
#pragma unroll
          for (int ct = 0; ct < 4; ++ct)
            bn[ct] = *(const v2f*)(bSub + (kc + 1) * BK_STRIDE + ct * 32);
        }
#pragma unroll
        for (int ct = 0; ct < 4; ++ct) {
          acc[ct] = __builtin_amdgcn_wmma_f32_16x16x4_f32(
              false, a0[kc], false, bf[ct], (short)0, acc[ct], false, false);
          acc[4 + ct] = __builtin_amdgcn_wmma_f32_16x16x4_f32(
              false, a1[kc], false, bf[ct], (short)0, acc[4 + ct], false, false);
        }
        if (kc < 7) {
#pragma unroll
          for (int ct = 0; ct < 4; ++ct) bf[ct] = bn[ct];
        }
      }
    }
  }

  // D layout: VGPR v, lanes 0-15 -> M=v, N=lane; lanes 16-31 -> M=v+8, N=lane-16
  const int rowOff = (lane < 16) ? 0 : 8;
  if (m0 + 32 <= N) {
    float* outBase =
        xt + (size_t)(m0 + rowOff) * XT_STRIDE + r * RGCN_H + c0 + halfLane;
#pragma unroll
    for (int a = 0; a < 2; ++a) {
#pragma unroll
      for (int ct = 0; ct < 4; ++ct) {
#pragma unroll
        for (int v = 0; v < 8; ++v) {
          outBase[(size_t)(a * 16 + v) * XT_STRIDE + ct * 16] = acc[a * 4 + ct][v];
        }
      }
    }
  } else {
#pragma unroll
    for (int a = 0; a < 2; ++a) {
#pragma unroll
      for (int ct = 0; ct < 4; ++ct) {
#pragma unroll
        for (int v = 0; v < 8; ++v) {
          const int row = m0 + a * 16 + v + rowOff;
          if (row < N)
            xt[(size_t)row * XT_STRIDE + r * RGCN_H + c0 + ct * 16 + halfLane] =
                acc[a * 4 + ct][v];
        }
      }
    }
  }
}

// ---------------------------------------------------------------------------
// Edge scatter: one wave per edge. msg = xt[src, et, :] * (1/deg(dst,et));
// atomically accumulate into xt[dst, R, :] (which already holds X @ root).
// ---------------------------------------------------------------------------
__global__ __launch_bounds__(256) void k_edge_scatter(const int* __restrict__ src,
                                                      const int* __restrict__ dst,
                                                      const int* __restrict__ et,
                                                      const float* __restrict__ norm,
                                                      float* __restrict__ xt, int E) {
  const int lane = threadIdx.x & 31;
  const int e = (blockIdx.x * 256 + threadIdx.x) >> 5;
  if (e >= E) return;
  const int s = src[e];
  const int d = dst[e];
  const int t = et[e];
  const float nv = norm[d * RGCN_R + t];

  const v4f v = *(const v4f*)(xt + (size_t)s * XT_STRIDE + t * RGCN_H + lane * 4);
  float* o = xt + (size_t)d * XT_STRIDE + RGCN_R * RGCN_H + lane * 4;
  unsafeAtomicAdd(o + 0, v.x * nv);
  unsafeAtomicAdd(o + 1, v.y * nv);
  unsafeAtomicAdd(o + 2, v.z * nv);
  unsafeAtomicAdd(o + 3, v.w * nv);
}

// ---------------------------------------------------------------------------
// out[n, :] = relu(xt[n, R, :] + bias)
// ---------------------------------------------------------------------------
__global__ __launch_bounds__(256) void k_bias_relu(const float* __restrict__ xt,
                                                   const float* __restrict__ bias,
                                                   float* __restrict__ out, int N) {
  int i = blockIdx.x * 256 + threadIdx.x;  // over N * 32 float4 groups
  if (i >= N * (RGCN_H / 4)) return;
  const int n = i >> 5;
  const int h = (i & 31) * 4;
  v4f v = *(const v4f*)(xt + (size_t)n * XT_STRIDE + RGCN_R * RGCN_H + h);
  v4f b = *(const v4f*)(bias + h);
  v4f o;
  o.x = fmaxf(v.x + b.x, 0.0f);
  o.y = fmaxf(v.y + b.y, 0.0f);
  o.z = fmaxf(v.z + b.z, 0.0f);
  o.w = fmaxf(v.w + b.w, 0.0f);
  *(v4f*)(out + (size_t)n * RGCN_H + h) = o;
}

// ---------------------------------------------------------------------------
// Final linear: out[n, 0:16] = concat(e1,e2,e3)[n, :] @ LW + LB.
// One wave per 16 nodes; single 16-col tile; K = 3 x 128 on WMMA f32 16x16x4.
// LW (24 KB) stays L2/L0 resident.  (~0.5% of total FLOPs.)
// ---------------------------------------------------------------------------
__global__ __launch_bounds__(256) void k_final(const float* __restrict__ e1,
                                               const float* __restrict__ e2,
                                               const float* __restrict__ e3,
                                               const float* __restrict__ LW,
                                               const float* __restrict__ LB,
                                               float* __restrict__ out, int N) {
  const int tid = threadIdx.x;
  const int lane = tid & 31;
  const int wave = tid >> 5;
  const int halfLane = lane & 15;
  const int kSel = (lane >> 4) << 1;
  const int n0 = (blockIdx.x * 8 + wave) * 16;

  int aRow = n0 + halfLane;
  if (aRow >= N) aRow = N - 1;

  v8f acc = v8f{};
  const float* segs[3] = {e1, e2, e3};
#pragma unroll
  for (int s = 0; s < 3; ++s) {
    const float* A = segs[s] + (size_t)aRow * RGCN_H;
    const float* B = LW + (size_t)(s * RGCN_H) * RGCN_C;
    // Prefetch the segment's 32 A fragments.
    v2f afrag[32];
#pragma unroll
    for (int kc = 0; kc < 32; ++kc)
      afrag[kc] = *(const v2f*)(A + kc * 4 + kSel);
#pragma unroll
    for (int kc = 0; kc < 32; ++kc) {
      const int kb = kc * 4 + kSel;
      v2f b;
      b.x = B[kb * RGCN_C + halfLane];
      b.y = B[(kb + 1) * RGCN_C + halfLane];
      acc = __builtin_amdgcn_wmma_f32_16x16x4_f32(
          false, afrag[kc], false, b, (short)0, acc, false, false);
    }
  }

  const int rowOff = (lane < 16) ? 0 : 8;
  const float bb = LB[halfLane];
#pragma unroll
  for (int v = 0; v < 8; ++v) {
    const int row = n0 + v + rowOff;
    if (row < N) out[(size_t)row * RGCN_C + halfLane] = acc[v] + bb;
  }
}

// ---------------------------------------------------------------------------
// Driver
// ---------------------------------------------------------------------------
extern "C" void kernel_launch(void* const* d_in, const int* in_sizes, int n_in,
                              void* d_out, int out_size, void* d_ws, size_t ws_size,
                              hipStream_t stream) {
  const float* x     = (const float*)d_in[0];
  const int*   ei    = (const int*)d_in[1];
  const int*   et    = (const int*)d_in[2];
  const float* w1    = (const float*)d_in[3];
  const float* root1 = (const float*)d_in[4];
  const float* b1    = (const float*)d_in[5];
  const float* w2    = (const float*)d_in[6];
  const float* root2 = (const float*)d_in[7];
  const float* b2    = (const float*)d_in[8];
  const float* w3    = (const float*)d_in[9];
  const float* root3 = (const float*)d_in[10];
  const float* b3    = (const float*)d_in[11];
  const float* lw    = (const float*)d_in[12];
  const float* lb    = (const float*)d_in[13];

  const int N = in_sizes[0] / RGCN_D;
  const int E = in_sizes[2];
  const int* src = ei;
  const int* dstp = ei + E;

  // Workspace layout (floats): xt [N*1152] | out1 | out2 | out3 [N*128 each]
  //                            | cnt [N*8 ints] | norm [N*8 floats]
  float* xt   = (float*)d_ws;
  float* out1 = xt + (size_t)N * XT_STRIDE;
  float* out2 = out1 + (size_t)N * RGCN_H;
  float* out3 = out2 + (size_t)N * RGCN_H;
  int*   cnt  = (int*)(out3 + (size_t)N * RGCN_H);
  float* nrm  = (float*)(cnt + (size_t)N * RGCN_R);

  const int NR = N * RGCN_R;
  k_zero_cnt<<<(NR + 255) / 256, 256, 0, stream>>>(cnt, NR);
  k_count<<<(E + 255) / 256, 256, 0, stream>>>(dstp, et, cnt, E);
  k_norm<<<(NR + 255) / 256, 256, 0, stream>>>(cnt, nrm, NR);

  dim3 gg((N + 127) / 128, RGCN_R + 1);
  const int ebl = (E + 7) / 8;            // one wave per edge, 8 waves/block
  const int nbl = (N * 32 + 255) / 256;   // bias+relu float4 groups
  const int fbl = (N + 127) / 128;        // final linear: 8 waves x 16 rows

  // Layer 1
  k_gemm_rel<<<gg, 256, 0, stream>>>(x, w1, root1, xt, N);
  k_edge_scatter<<<ebl, 256, 0, stream>>>(src, dstp, et, nrm, xt, E);
  k_bias_relu<<<nbl, 256, 0, stream>>>(xt, b1, out1, N);

  // Layer 2
  k_gemm_rel<<<gg, 256, 0, stream>>>(out1, w2, root2, xt, N);
  k_edge_scatter<<<ebl, 256, 0, stream>>>(src, dstp, et, nrm, xt, E);
  k_bias_relu<<<nbl, 256, 0, stream>>>(xt, b2, out2, N);

  // Layer 3
  k_gemm_rel<<<gg, 256, 0, stream>>>(out2, w3, root3, xt, N);
  k_edge_scatter<<<ebl, 256, 0, stream>>>(src, dstp, et, nrm, xt, E);
  k_bias_relu<<<nbl, 256, 0, stream>>>(xt, b3, out3, N);

  // Final linear
  k_final<<<fbl, 256, 0, stream>>>(out1, out2, out3, lw, lb, (float*)d_out, N);
}